// MetaHyperNetwork_20830591385783
// MI455X (gfx1250) — compile-verified
//
#include <hip/hip_runtime.h>

// ---------------------------------------------------------------------------
// MetaHyperNetwork for MI455X (gfx1250), single-wave wave32 kernel.
//
//   sim  = hw . hw_emb[e] / sqrt(10)             (e = 0..49)
//   gate = softmax(sin(sim))
//   idx  = round(x * 100)
//   out[d] = sum_e gate[e] * expert_emb[e, idx, d]   (d = 0..29)
//
// Final contraction on the matrix pipe:
//   out^T (30x1) = rows^T (30x50) x gate (50x1)
// via V_WMMA_F32_16X16X4_F32 (f32 A/B keeps full reference precision):
//   - M = d: two tiles (0..15, 16..31 zero-padded)
//   - K = e: 13 steps of 4 (zero-padded 50 -> 52)
//   - B = gate broadcast across all N columns -> every D column equals out
// All zero-padding lives in LDS so operand fetches are unconditional
// ds_load_b64's, clustered before one s_wait_dscnt and 26 chained WMMAs.
// ---------------------------------------------------------------------------

typedef float v2f __attribute__((ext_vector_type(2)));
typedef float v8f __attribute__((ext_vector_type(8)));

#define NUM_INTERVALS 101
#define HW_DIM        10
#define NUM_E         50
#define OUT_D         30
#define E_PAD         52            // K padded to 13 * 4
#define E_STRIDE      58            // even (keeps b64 alignment), conflict-friendly
#define NSTEP         13

__global__ __launch_bounds__(32)
void mhn_wmma_kernel(const float* __restrict__ x,
                     const float* __restrict__ hw,
                     const float* __restrict__ hw_emb,
                     const float* __restrict__ expert_emb,
                     float* __restrict__ out)
{
    // s_rowsT[d][e]: d in 0..31 (30,31 zero), e in 0..51 (50,51 zero)
    __shared__ __align__(16) float s_rowsT[32 * E_STRIDE];
    __shared__ __align__(16) float s_gate[64];          // zero beyond e=49

    const int lane = threadIdx.x;          // 0..31, single wave32

    // ---- interval index (rintf == round-half-to-even == jnp.round) ---------
    const int idx = (int)rintf(x[0] * (float)(NUM_INTERVALS - 1));

    // ---- hw into registers (broadcast loads) -------------------------------
    float hwr[HW_DIM];
    #pragma unroll
    for (int i = 0; i < HW_DIM; ++i) hwr[i] = hw[i];

    const float inv_sqrt10 = 0.31622776601683794f;

    // ---- exp(sin(sim)) : lane owns e = lane (always valid) and e = lane+32 -
    float ex0, ex1 = 0.0f;
    {
        float s = 0.0f;
        #pragma unroll
        for (int i = 0; i < HW_DIM; ++i)
            s = fmaf(hwr[i], hw_emb[lane * HW_DIM + i], s);
        ex0 = expf(sinf(s * inv_sqrt10));
    }
    if (lane + 32 < NUM_E) {
        float s = 0.0f;
        #pragma unroll
        for (int i = 0; i < HW_DIM; ++i)
            s = fmaf(hwr[i], hw_emb[(lane + 32) * HW_DIM + i], s);
        ex1 = expf(sinf(s * inv_sqrt10));
    }

    // ---- softmax denominator: wave32 butterfly reduction -------------------
    float total = ex0 + ex1;
    #pragma unroll
    for (int off = 16; off >= 1; off >>= 1)
        total += __shfl_xor(total, off, 32);
    const float invT = 1.0f / total;

    s_gate[lane]      = ex0 * invT;
    s_gate[lane + 32] = (lane + 32 < NUM_E) ? ex1 * invT : 0.0f;

    // ---- stage rows transposed: s_rowsT[d][e], coalesced global reads ------
    const float* row_base = expert_emb + (size_t)idx * OUT_D;
    if (lane < OUT_D) {
        for (int e = 0; e < NUM_E; ++e)
            s_rowsT[lane * E_STRIDE + e] =
                row_base[(size_t)e * (NUM_INTERVALS * OUT_D) + lane];
        s_rowsT[lane * E_STRIDE + 50] = 0.0f;
        s_rowsT[lane * E_STRIDE + 51] = 0.0f;
    } else {                                   // d = 30, 31: zero rows
        #pragma unroll
        for (int e = 0; e < E_PAD; ++e)
            s_rowsT[lane * E_STRIDE + e] = 0.0f;
    }

    __syncthreads();

    // ---- gather all WMMA operands (unconditional ds_load_b64's) ------------
    // 32-bit A 16x4 layout: lane l, VGPR v  <->  A[l&15][2*(l>>4) + v]
    // 32-bit B 4x16 layout: lane l, VGPR v  <->  B[2*(l>>4) + v][l&15]
    const int hi = lane >> 4;   // 0 | 1
    const int mn = lane & 15;   // M (A/D rows) and N (B cols)

    v2f A0[NSTEP], A1[NSTEP], B[NSTEP];
    #pragma unroll
    for (int s = 0; s < NSTEP; ++s) {
        const int e0 = 4 * s + 2 * hi;                       // even -> 8B aligned
        A0[s] = *(const v2f*)&s_rowsT[mn * E_STRIDE + e0];         // d = mn
        A1[s] = *(const v2f*)&s_rowsT[(mn + 16) * E_STRIDE + e0];  // d = mn+16
        B[s]  = *(const v2f*)&s_gate[e0];                          // same for all N
    }

    // ---- 26 chained WMMAs, two interleaved accumulators --------------------
    v8f c0 = {};   // d = 0..15
    v8f c1 = {};   // d = 16..31 (rows 30,31 stay zero)
    #pragma unroll
    for (int s = 0; s < NSTEP; ++s) {
        // 8 args: (neg_a, A, neg_b, B, c_mod, C, reuse_a, reuse_b)
        c0 = __builtin_amdgcn_wmma_f32_16x16x4_f32(false, A0[s], false, B[s],
                                                   (short)0, c0, false, false);
        c1 = __builtin_amdgcn_wmma_f32_16x16x4_f32(false, A1[s], false, B[s],
                                                   (short)0, c1, false, false);
    }

    // ---- writeout ----------------------------------------------------------
    // D layout: VGPR v holds M=v (lanes 0-15) / M=v+8 (lanes 16-31); all N
    // columns identical by construction, so lanes 0 and 16 carry everything.
    if (mn == 0) {
        const int dbase = hi * 8;
        #pragma unroll
        for (int v = 0; v < 8; ++v) {
            out[dbase + v] = c0[v];            // d = 0..15
            int d1 = 16 + dbase + v;           // d = 16..29
            if (d1 < OUT_D) out[d1] = c1[v];
        }
    }
}

extern "C" void kernel_launch(void* const* d_in, const int* in_sizes, int n_in,
                              void* d_out, int out_size, void* d_ws, size_t ws_size,
                              hipStream_t stream) {
    const float* x          = (const float*)d_in[0];  // (1,1)
    const float* hw         = (const float*)d_in[1];  // (10,)
    const float* hw_emb     = (const float*)d_in[2];  // (50,10)
    const float* expert_emb = (const float*)d_in[3];  // (50,101,30)
    float* out              = (float*)d_out;          // (6,5) = 30 floats

    mhn_wmma_kernel<<<1, 32, 0, stream>>>(x, hw, hw_emb, expert_emb, out);
}